// RotMat_19988777796042
// MI455X (gfx1250) — compile-verified
//
#include <hip/hip_runtime.h>
#include <math.h>

// N = 4096 Givens-rotation composition (tournament schedule).
// Each workgroup owns an N x 8 fp32 column slab of U in LDS (two float4
// "planes" of 4096 rows = 128 KB), initialized to the identity slab, and
// applies all 4095 rounds in LDS (1 barrier per round). Schedule records
// {i,j,cos,sin} are precomputed once (gather + sincos) into d_ws and
// prefetched one round ahead into THREAD-PRIVATE LDS staging slots using
// CDNA5 async global->LDS b128 loads, synchronized per-wave with
// s_wait_asynccnt / s_wait_dscnt (no cross-wave staging barriers needed).
// Total LDS = 160 KB -> two workgroups (64 waves) co-reside per 320 KB WGP.

#define NM    4096
#define RND   (NM - 1)    // 4095 rounds
#define PAIRS (NM / 2)    // 2048 pairs per round
#define TPB   1024        // 32 wave32 waves
#define COLS  8           // columns per workgroup

// ---------------------------------------------------------------------------
// Precompute packed records: recs[r*PAIRS + p] = {i, j, cos(th), sin(th)}
// ---------------------------------------------------------------------------
__global__ void rot_precompute(const float* __restrict__ thetas,
                               const int*  __restrict__ ri,
                               const int*  __restrict__ rj,
                               const int*  __restrict__ rt,
                               int4* __restrict__ recs, int total) {
    int idx = blockIdx.x * blockDim.x + threadIdx.x;
    if (idx >= total) return;
    float th = thetas[rt[idx]];
    float s, c;
    sincosf(th, &s, &c);
    int4 r;
    r.x = ri[idx];
    r.y = rj[idx];
    r.z = __float_as_int(c);
    r.w = __float_as_int(s);
    recs[idx] = r;
}

// ---------------------------------------------------------------------------
// Rotation helpers (fp32, fma form)
// ---------------------------------------------------------------------------
__device__ __forceinline__ float4 rot_a(float c, float s, float4 u, float4 v) {
    float4 r;
    r.x = fmaf(c, u.x, -s * v.x);
    r.y = fmaf(c, u.y, -s * v.y);
    r.z = fmaf(c, u.z, -s * v.z);
    r.w = fmaf(c, u.w, -s * v.w);
    return r;
}
__device__ __forceinline__ float4 rot_b(float c, float s, float4 u, float4 v) {
    float4 r;
    r.x = fmaf(s, u.x, c * v.x);
    r.y = fmaf(s, u.y, c * v.y);
    r.z = fmaf(s, u.z, c * v.z);
    r.w = fmaf(s, u.w, c * v.w);
    return r;
}

// ---------------------------------------------------------------------------
// Main kernel: one workgroup per 8-column slab, all rounds in LDS.
// ---------------------------------------------------------------------------
__global__ __launch_bounds__(TPB)
void rot_apply(const float* __restrict__ thetas,
               const int*  __restrict__ ri,
               const int*  __restrict__ rj,
               const int*  __restrict__ rt,
               const int4* __restrict__ recs,
               int use_recs,
               float* __restrict__ out) {
    // plane[g][row] holds cols [c0+4g, c0+4g+4) of row `row`.
    // Fixed g + random row -> b128 bank start (4*row)%64 covers all 64 banks.
    __shared__ float4 plane[2][NM];      // 128 KB
    __shared__ int4   stage[PAIRS];      // 32 KB, slot p owned by thread p%TPB

    const int t  = threadIdx.x;
    const int c0 = blockIdx.x * COLS;
    const int p0 = t;                    // this thread's two pairs per round
    const int p1 = t + TPB;

    // ---- identity slab init ----
    for (int idx = t; idx < 2 * NM; idx += TPB) {
        int g   = idx >> 12;             // idx / 4096
        int row = idx & (NM - 1);
        float4 v = make_float4(0.f, 0.f, 0.f, 0.f);
        int d = row - (c0 + g * 4);
        if (d == 0) v.x = 1.f;
        else if (d == 1) v.y = 1.f;
        else if (d == 2) v.z = 1.f;
        else if (d == 3) v.w = 1.f;
        plane[g][row] = v;
    }

    if (use_recs) {
        const unsigned lds0 = (unsigned)(uintptr_t)(&stage[p0]);
        const unsigned lds1 = (unsigned)(uintptr_t)(&stage[p1]);

        // Prologue: async-stage round 0 records into my private slots.
        asm volatile("global_load_async_to_lds_b128 %0, %1, %2"
                     :: "v"(lds0), "v"((unsigned)(p0 * 16)), "s"(recs) : "memory");
        asm volatile("global_load_async_to_lds_b128 %0, %1, %2"
                     :: "v"(lds1), "v"((unsigned)(p1 * 16)), "s"(recs) : "memory");
        __syncthreads();                 // identity init complete

        for (int r = 0; r < RND; ++r) {
            // My round-r records are staged; pull them to registers.
            asm volatile("s_wait_asynccnt 0" ::: "memory");
            int4 ra = stage[p0];
            int4 rb = stage[p1];

            // Prefetch round r+1 into the same private slots. Only ordering
            // needed: my ds reads above must complete first (s_wait_dscnt).
            if (r + 1 < RND) {
                unsigned base = (unsigned)(r + 1) * (unsigned)(PAIRS * 16);
                asm volatile("s_wait_dscnt 0" ::: "memory");
                asm volatile("global_load_async_to_lds_b128 %0, %1, %2"
                             :: "v"(lds0), "v"(base + (unsigned)(p0 * 16)), "s"(recs) : "memory");
                asm volatile("global_load_async_to_lds_b128 %0, %1, %2"
                             :: "v"(lds1), "v"(base + (unsigned)(p1 * 16)), "s"(recs) : "memory");
            }

            // Apply my two disjoint rotations to the slab (pure LDS traffic).
            {
                int i = ra.x, j = ra.y;
                float c = __int_as_float(ra.z), s = __int_as_float(ra.w);
                float4 ui0 = plane[0][i], uj0 = plane[0][j];
                float4 ui1 = plane[1][i], uj1 = plane[1][j];
                plane[0][i] = rot_a(c, s, ui0, uj0);
                plane[1][i] = rot_a(c, s, ui1, uj1);
                plane[0][j] = rot_b(c, s, ui0, uj0);
                plane[1][j] = rot_b(c, s, ui1, uj1);
            }
            {
                int i = rb.x, j = rb.y;
                float c = __int_as_float(rb.z), s = __int_as_float(rb.w);
                float4 ui0 = plane[0][i], uj0 = plane[0][j];
                float4 ui1 = plane[1][i], uj1 = plane[1][j];
                plane[0][i] = rot_a(c, s, ui0, uj0);
                plane[1][i] = rot_a(c, s, ui1, uj1);
                plane[0][j] = rot_b(c, s, ui0, uj0);
                plane[1][j] = rot_b(c, s, ui1, uj1);
            }

            __syncthreads();             // single per-round barrier
        }
    } else {
        // Fallback (ws too small): gather schedule + sincos in-kernel.
        __syncthreads();
        for (int r = 0; r < RND; ++r) {
            int base = r * PAIRS;
            #pragma unroll
            for (int k = 0; k < 2; ++k) {
                int p = base + t + k * TPB;
                int i = ri[p], j = rj[p];
                float th = thetas[rt[p]];
                float s, c;
                sincosf(th, &s, &c);
                float4 ui0 = plane[0][i], uj0 = plane[0][j];
                float4 ui1 = plane[1][i], uj1 = plane[1][j];
                plane[0][i] = rot_a(c, s, ui0, uj0);
                plane[1][i] = rot_a(c, s, ui1, uj1);
                plane[0][j] = rot_b(c, s, ui0, uj0);
                plane[1][j] = rot_b(c, s, ui1, uj1);
            }
            __syncthreads();
        }
    }

    // ---- write slab to row-major output ----
    for (int idx = t; idx < 2 * NM; idx += TPB) {
        int g   = idx >> 12;
        int row = idx & (NM - 1);
        float4 v = plane[g][row];
        *(float4*)(out + (size_t)row * NM + c0 + g * 4) = v;
    }
}

// ---------------------------------------------------------------------------
// Host entry
// ---------------------------------------------------------------------------
extern "C" void kernel_launch(void* const* d_in, const int* in_sizes, int n_in,
                              void* d_out, int out_size, void* d_ws, size_t ws_size,
                              hipStream_t stream) {
    const float* thetas = (const float*)d_in[0];
    const int*   ri     = (const int*)d_in[1];
    const int*   rj     = (const int*)d_in[2];
    const int*   rt     = (const int*)d_in[3];
    float*       out    = (float*)d_out;

    // Derive N from n_thetas = N(N-1)/2; this build is specialized for 4096.
    double T = (double)in_sizes[0];
    int n = (int)((1.0 + sqrt(1.0 + 8.0 * T)) / 2.0 + 0.5);
    if (n != NM) return;

    const int    total = RND * PAIRS;
    const size_t need  = (size_t)total * sizeof(int4);   // ~128 MB
    const int use_recs = (d_ws != nullptr && ws_size >= need) ? 1 : 0;
    int4* recs = (int4*)d_ws;

    if (use_recs) {
        rot_precompute<<<(total + 255) / 256, 256, 0, stream>>>(
            thetas, ri, rj, rt, recs, total);
    }
    rot_apply<<<NM / COLS, TPB, 0, stream>>>(
        thetas, ri, rj, rt, recs, use_recs, out);
}